// QuantumMLModel_55980603736430
// MI455X (gfx1250) — compile-verified
//
#include <hip/hip_runtime.h>
#include <math.h>

typedef float v2f __attribute__((ext_vector_type(2)));
typedef float v8f __attribute__((ext_vector_type(8)));

#define NQ      8
#define DIM     256      // 2^NQ
#define NL      4
#define NC      3
#define BATCH   65536
#define MPAD    260      // 256 + 4 pad -> conflict-free ds reads
#define TILE_F  (16 * MPAD)

// ---------------------------------------------------------------------------
// Kernel 1: propagate each basis column through the (batch-independent) layer
// circuit, fold the (-i)^popcount embedding phase, and emit the separate real
// planes packR[i][j], packI[i][j] plus the folded head g[k][i].
// ---------------------------------------------------------------------------
__global__ __launch_bounds__(DIM)
void build_circuit_matrices(const float* __restrict__ weights,   // [NL][NQ][4]
                            const float* __restrict__ fc_w,      // [NC][NQ]
                            float* __restrict__ packR,           // [DIM][DIM]
                            float* __restrict__ packI,           // [DIM][DIM]
                            float* __restrict__ g)               // [NC][DIM]
{
    __shared__ float sr[DIM];
    __shared__ float si[DIM];
    const int t   = threadIdx.x;
    const int col = blockIdx.x;

    if (col == 0) {
        // g[k][i] = sum_w fc_w[k][w] * (bit(i, NQ-1-w) ? -1 : +1)
        #pragma unroll
        for (int k = 0; k < NC; ++k) {
            float acc = 0.f;
            #pragma unroll
            for (int w = 0; w < NQ; ++w) {
                float sgn = ((t >> (NQ - 1 - w)) & 1) ? -1.f : 1.f;
                acc += fc_w[k * NQ + w] * sgn;
            }
            g[k * DIM + t] = acc;
        }
    }

    sr[t] = (t == col) ? 1.f : 0.f;
    si[t] = 0.f;
    __syncthreads();

    for (int layer = 0; layer < NL; ++layer) {
        // Rot(phi, theta, omega) on each qubit
        for (int q = 0; q < NQ; ++q) {
            const float* wp = weights + (layer * NQ + q) * 4;
            float phi = wp[0], theta = wp[1], omega = wp[2];
            float ct, st, epr, epi, emr, emi;
            sincosf(0.5f * theta, &st, &ct);
            sincosf(-0.5f * (phi + omega), &epi, &epr);   // ep = exp(-i(phi+omega)/2)
            sincosf(-0.5f * (phi - omega), &emi, &emr);   // em = exp(-i(phi-omega)/2)
            float u00r =  epr * ct, u00i =  epi * ct;     // ep*ct
            float u01r = -emr * st, u01i =  emi * st;     // -conj(em)*st
            float u10r =  emr * st, u10i =  emi * st;     // em*st
            float u11r =  epr * ct, u11i = -epi * ct;     // conj(ep)*ct

            const int mask = 1 << (NQ - 1 - q);
            const int p = t ^ mask;
            float ar = sr[t], ai = si[t];
            float br = sr[p], bi = si[p];
            __syncthreads();
            float nr, ni;
            if ((t & mask) == 0) {        // I hold the bit=0 amplitude
                nr = u00r * ar - u00i * ai + u01r * br - u01i * bi;
                ni = u00r * ai + u00i * ar + u01r * bi + u01i * br;
            } else {                      // partner holds the bit=0 amplitude
                nr = u10r * br - u10i * bi + u11r * ar - u11i * ai;
                ni = u10r * bi + u10i * br + u11r * ai + u11i * ar;
            }
            sr[t] = nr; si[t] = ni;
            __syncthreads();
        }
        // CNOT chain: control q, target q+1
        for (int q = 0; q < NQ - 1; ++q) {
            const int mc = 1 << (NQ - 1 - q);
            const int mt = mc >> 1;
            const int src = (t & mc) ? (t ^ mt) : t;
            float vr = sr[src], vi = si[src];
            __syncthreads();
            sr[t] = vr; si[t] = vi;
            __syncthreads();
        }
    }

    // Fold embedding phase (-i)^popcount(col) into column col of W.
    const int k4 = __popc(col) & 3;
    const float rrv = (k4 == 0) ? 1.f : (k4 == 2) ? -1.f : 0.f;
    const float riv = (k4 == 1) ? -1.f : (k4 == 3) ? 1.f : 0.f;
    const float wr = sr[t], wi = si[t];
    packR[t * DIM + col] = wr * rrv - wi * riv;   // Br(i=t, j=col)
    packI[t * DIM + col] = wr * riv + wi * rrv;   // Bi(i=t, j=col)
}

// ---------------------------------------------------------------------------
// Kernel 2: 4 waves/block, each wave owns 16 batch rows.  The block stages
// each 16-state B-tile (both planes) into double-buffered LDS with
// GLOBAL_LOAD_ASYNC_TO_LDS_B128 (shared by all 4 waves), while each wave's
// m(b,j) table also lives in LDS.  Inner loop: 2x v_wmma_f32_16x16x4_f32 per
// K-chunk, accumulating real/imag amplitudes in f32.
//
// LDS layout: stage buffers FIRST so all their byte offsets stay below the
// 16-bit DS offset-field limit (the compiler can then fold buffer/plane/chunk
// constants into the ds_load offset instead of VALU address adds); the
// per-wave m tables sit above and are reached via a per-wave base register
// with small 2addr offsets.
// ---------------------------------------------------------------------------
__global__ __launch_bounds__(128)
void qml_forward(const float* __restrict__ x,      // [BATCH][NQ]
                 const float* __restrict__ fc_b,   // [NC]
                 const float* __restrict__ packR,  // [DIM][DIM]
                 const float* __restrict__ packI,  // [DIM][DIM]
                 const float* __restrict__ g,      // [NC][DIM]
                 float* __restrict__ out)          // [BATCH][NC]
{
    // [0, 8*TILE_F)            : stage buffers: (buf*2 + plane)*TILE_F
    // [8*TILE_F, 12*TILE_F)    : per-wave m tables
    __shared__ float smem[12 * TILE_F];
    float* stage = smem;
    float* mtab  = smem + 8 * TILE_F;

    const int tid  = threadIdx.x;
    const int lane = tid & 31;
    const int wave = tid >> 5;
    const int row  = lane & 15;     // A-matrix M index / B & D N index
    const int hi   = lane >> 4;     // 0: K pair {0,1}; 1: K pair {2,3}
    const int koff = hi << 1;
    const int batchBase = (blockIdx.x * 4 + wave) * 16;
    const int b = batchBase + row;

    // Async-stage one 16-row B tile (both planes) into LDS buffer bufsel.
    // 2048 x 16B segments / 128 threads = 16 async ops per thread (per wave:
    // 16 wave-instructions -> ASYNCcnt +16 per tile, completing in order).
    auto stageTile = [&](int itile, int bufsel) {
        #pragma unroll
        for (int it = 0; it < 16; ++it) {
            const int seg  = it * 128 + tid;        // 0..2047
            const int pl   = seg >> 10;             // plane: 0=R, 1=I
            const int s    = seg & 1023;
            const int r    = s >> 6;                // tile row 0..15
            const int jseg = s & 63;                // 16B segment in row
            const float* gp = (pl ? packI : packR) + (itile * 16 + r) * DIM + jseg * 4;
            float* lp = stage + (bufsel * 2 + pl) * TILE_F + r * MPAD + jseg * 4;
            const unsigned lds_off = (unsigned)(uintptr_t)lp;  // LDS aperture: addr[31:0]
            asm volatile("global_load_async_to_lds_b128 %0, %1, off"
                         :: "v"(lds_off), "v"(gp) : "memory");
        }
    };

    stageTile(0, 0);

    // cos/sin of half-angles, indexed by *bit position* p (wire = NQ-1-p)
    float cw[NQ], sw[NQ];
    #pragma unroll
    for (int p = 0; p < NQ; ++p)
        sincosf(0.5f * x[b * NQ + (NQ - 1 - p)], &sw[p], &cw[p]);

    // Fill m(b, j) = prod_p (bit(j,p) ? sw[p] : cw[p]); each lane does 128 j's.
    float* mbase = mtab + wave * TILE_F + row * MPAD;
    for (int jj = 0; jj < 128; ++jj) {
        const int j = (hi << 7) | jj;
        float v = ((j     ) & 1) ? sw[0] : cw[0];
        #pragma unroll
        for (int p = 1; p < NQ; ++p)
            v *= ((j >> p) & 1) ? sw[p] : cw[p];
        mbase[j] = v;
    }

    float po[NC][8];
    #pragma unroll
    for (int k = 0; k < NC; ++k)
        #pragma unroll
        for (int r = 0; r < 8; ++r) po[k][r] = 0.f;

    for (int itile = 0; itile < 16; ++itile) {
        const int cur = itile & 1;
        if (itile + 1 < 16) {
            stageTile(itile + 1, cur ^ 1);
            // <=16 outstanding -> tile 'itile' (issued earlier, in-order) done
            asm volatile("s_wait_asynccnt 0x10" ::: "memory");
        } else {
            asm volatile("s_wait_asynccnt 0x0" ::: "memory");
        }
        __syncthreads();   // tile visible to all 4 waves (also covers m-table on itile==0)

        const float* am = mbase + koff;
        const float* tR = stage + (cur * 2 + 0) * TILE_F + row * MPAD + koff;
        const float* tI = stage + (cur * 2 + 1) * TILE_F + row * MPAD + koff;

        v8f accR = {0.f, 0.f, 0.f, 0.f, 0.f, 0.f, 0.f, 0.f};
        v8f accI = {0.f, 0.f, 0.f, 0.f, 0.f, 0.f, 0.f, 0.f};

        for (int c = 0; c < 64; ++c) {
            v2f av = *(const v2f*)(am + 4 * c);   // A frag: m(b, 4c + {koff,koff+1})
            v2f bR = *(const v2f*)(tR + 4 * c);   // B frag: Br(iN, 4c + {koff,koff+1})
            v2f bI = *(const v2f*)(tI + 4 * c);
            accR = __builtin_amdgcn_wmma_f32_16x16x4_f32(
                       false, av, false, bR, (short)0, accR, false, false);
            accI = __builtin_amdgcn_wmma_f32_16x16x4_f32(
                       false, av, false, bI, (short)0, accI, false, false);
        }

        const int iN = itile * 16 + row;          // state index of my N column
        const float g0 = g[iN], g1 = g[DIM + iN], g2 = g[2 * DIM + iN];
        #pragma unroll
        for (int r = 0; r < 8; ++r) {
            const float p = accR[r] * accR[r] + accI[r] * accI[r];
            po[0][r] += p * g0;
            po[1][r] += p * g1;
            po[2][r] += p * g2;
        }
        __syncthreads();   // all waves done reading buf[cur] before it is restaged
    }

    // Reduce over the 16 lanes of each half (the N dimension).
    #pragma unroll
    for (int k = 0; k < NC; ++k)
        #pragma unroll
        for (int r = 0; r < 8; ++r) {
            float v = po[k][r];
            v += __shfl_xor(v, 1, 32);
            v += __shfl_xor(v, 2, 32);
            v += __shfl_xor(v, 4, 32);
            v += __shfl_xor(v, 8, 32);
            po[k][r] = v;
        }

    if (row == 0) {
        const float b0 = fc_b[0], b1 = fc_b[1], b2 = fc_b[2];
        #pragma unroll
        for (int r = 0; r < 8; ++r) {
            const int M  = r + hi * 8;            // D layout: M = vgpr + 8*half
            const int ob = (batchBase + M) * NC;
            out[ob + 0] = po[0][r] + b0;
            out[ob + 1] = po[1][r] + b1;
            out[ob + 2] = po[2][r] + b2;
        }
    }
}

// ---------------------------------------------------------------------------
extern "C" void kernel_launch(void* const* d_in, const int* in_sizes, int n_in,
                              void* d_out, int out_size, void* d_ws, size_t ws_size,
                              hipStream_t stream) {
    const float* x       = (const float*)d_in[0];   // [BATCH][NQ]
    const float* weights = (const float*)d_in[1];   // [NL][NQ][4]
    const float* fc_w    = (const float*)d_in[2];   // [NC][NQ]
    const float* fc_b    = (const float*)d_in[3];   // [NC]
    float* out  = (float*)d_out;

    float* packR = (float*)d_ws;                    // 256*256 floats = 256 KB
    float* packI = packR + DIM * DIM;               // 256 KB
    float* g     = packI + DIM * DIM;               // 3*256 floats

    build_circuit_matrices<<<DIM, DIM, 0, stream>>>(weights, fc_w, packR, packI, g);
    qml_forward<<<BATCH / 64, 128, 0, stream>>>(x, fc_b, packR, packI, g, out);
}